// PoseAwareAdversarialPatch_62749472194983
// MI455X (gfx1250) — compile-verified
//
#include <hip/hip_runtime.h>
#include <math.h>
#include <stdint.h>

// ---------------------------------------------------------------------------
// PoseAwareAdversarialPatch — MI455X (gfx1250) single-pass layered composite.
//
// Roofline: ~12.6 MB of HBM traffic total (read 2x3x512x512 f32, write same)
// => ~0.54 us at 23.3 TB/s. Everything else (68-layer composite with per-tile
// pruning, bilinear of a 30x30 nearest-upsampled 3x3 texture) lives in
// registers/LDS. The 2.8 KB LUT (layer affine params + sigmoid texels) is
// DMA'd into LDS per workgroup with the CDNA5 Tensor Data Mover
// (tensor_load_to_lds + s_wait_tensorcnt).
// ---------------------------------------------------------------------------

#define BDIM   256
#define TILE   32
#define HW     512
#define KNUM   17
#define PNUM   4
#define BNUM   2
#define SNUM   (PNUM * KNUM)        // 68 layers per image, ordered s = p*17+k
#define NLAYER (BNUM * SNUM)        // 136
#define PARAM_FLOATS (NLAYER * 4)   // 544: {a, bx, by, mea} per layer
#define SIG_FLOATS   (KNUM * 9)     // 153: sigmoid(10*tex) 3x3 per keypoint
#define TBL_FLOATS   704            // padded (zero) so masked OOB reads are finite
#define IMG_PLANE (HW * HW)
#define IMG_ELEMS (BNUM * 3 * IMG_PLANE)

__device__ __constant__ float SR_LO[KNUM] = {
    0.10f,0.10f,0.10f,0.10f,0.10f, 0.15f,0.15f, 0.10f,0.10f,
    0.10f,0.10f, 0.20f,0.20f, 0.15f,0.15f, 0.10f,0.10f };
__device__ __constant__ float SR_HI[KNUM] = {
    0.20f,0.20f,0.20f,0.20f,0.20f, 0.30f,0.30f, 0.25f,0.25f,
    0.20f,0.20f, 0.40f,0.40f, 0.30f,0.30f, 0.25f,0.25f };

__device__ __forceinline__ float sigm(float x) { return 1.0f / (1.0f + expf(-x)); }

// ---------------------------------------------------------------------------
// prep: build the LUT in d_ws and emit alphas (2nd tuple output).
// Per layer (b,p,k): patch-space coord of image pixel X is ix = a*X + bx with
//   pps = max(scale_k * bbox_w, 3), a = 30/pps, bx = a*(0.5 - kx) + 14.5
// (algebraic fold of affine_grid(align_corners=False) + grid_sample coords).
// mea = sigmoid(alpha_logit_k) * [(v!=0) & (bbox_w>1)].
// ---------------------------------------------------------------------------
__global__ __launch_bounds__(BDIM) void pap_prep(
    const float* __restrict__ tkp,   // [B,P,K,3]
    const float* __restrict__ tbb,   // [B,P,4]
    const float* __restrict__ tex,   // [K,1,3,3]
    const float* __restrict__ slog,  // [K]
    const float* __restrict__ alog,  // [K]
    float* __restrict__ ws,          // LUT: 704 floats
    float* __restrict__ alphas_out)  // [K]
{
    const int t = threadIdx.x;
    if (t < KNUM) alphas_out[t] = sigm(alog[t]);

    if (t < NLAYER) {
        const int b = t / SNUM, s = t - b * SNUM;
        const int p = s / KNUM,  k = s - p * KNUM;
        const float alpha = sigm(alog[k]);
        const float scale = SR_LO[k] + sigm(slog[k]) * (SR_HI[k] - SR_LO[k]);
        const float* kpt = tkp + ((size_t)(b * PNUM + p) * KNUM + k) * 3;
        const float x = kpt[0], y = kpt[1], v = kpt[2];
        const float pw = tbb[(b * PNUM + p) * 4 + 2];
        const float pps = fmaxf(scale * pw, 3.0f);
        const float a = 30.0f / pps;
        const float mea = ((v != 0.0f) && (pw > 1.0f)) ? alpha : 0.0f;
        float* W4 = ws + t * 4;
        W4[0] = a;
        W4[1] = fmaf(a, 0.5f - x, 14.5f);
        W4[2] = fmaf(a, 0.5f - y, 14.5f);
        W4[3] = mea;
    }
    // sigmoid texels + zero padding (indices 544..703)
    if (t < TBL_FLOATS - PARAM_FLOATS) {
        float val = 0.0f;
        if (t < SIG_FLOATS) val = sigm(tex[t] * 10.0f);
        ws[PARAM_FLOATS + t] = val;
    }
}

// ---------------------------------------------------------------------------
// blend: one 32x32 tile per block; TDM the LUT to LDS; prune layers that miss
// the tile (order preserved == scan order); composite 4 px/thread.
// ---------------------------------------------------------------------------
#if defined(__HIP_DEVICE_COMPILE__) && \
    __has_builtin(__builtin_amdgcn_tensor_load_to_lds) && \
    __has_builtin(__builtin_amdgcn_s_wait_tensorcnt)
#define PAP_USE_TDM 1
#else
#define PAP_USE_TDM 0
#endif

__global__ __launch_bounds__(BDIM) void pap_blend(
    const float* __restrict__ img,   // [B,3,H,W]
    const float* __restrict__ ws,    // LUT: 704 floats
    float* __restrict__ out)         // [B,3,H,W]
{
    __shared__ float tbl[TBL_FLOATS];
    __shared__ int   plist[SNUM];
    __shared__ int   pcount;

    const int b  = blockIdx.z;
    const int X0 = blockIdx.x * TILE;
    const int Y0 = blockIdx.y * TILE;

#if PAP_USE_TDM
    if (threadIdx.x == 0) {
        typedef unsigned int v4u __attribute__((ext_vector_type(4)));
        typedef int v8i __attribute__((ext_vector_type(8)));
        typedef int v4i __attribute__((ext_vector_type(4)));
        const unsigned lds = (unsigned)(unsigned long long)(uintptr_t)tbl; // LDS byte offset
        const unsigned long long ga = (unsigned long long)(uintptr_t)ws;
        v4u g0;                                   // D# group0
        g0[0] = 1u;                               // count=1 (valid), user mode
        g0[1] = lds;                              // lds_addr [63:32]
        g0[2] = (unsigned)ga;                     // global_addr [95:64]
        g0[3] = (unsigned)((ga >> 32) & 0x1FFFFFFu) | 0x80000000u; // addr hi + type=2
        v8i g1;                                   // D# group1
        g1[0] = 0x00020000;                       // data_size = 4B; no mask/pad/iter
        g1[1] = (int)(TBL_FLOATS << 16);          // tensor_dim0[15:0]
        g1[2] = (1 << 16);                        // tensor_dim0 hi=0; tensor_dim1=1
        g1[3] = (int)(TBL_FLOATS << 16);          // tile_dim0 = 704
        g1[4] = 1;                                // tile_dim1 = 1; tile_dim2 unused
        g1[5] = TBL_FLOATS;                       // tensor_dim0_stride
        g1[6] = 0; g1[7] = 0;
        v4i gz4; gz4[0] = 0; gz4[1] = 0; gz4[2] = 0; gz4[3] = 0;
        v8i gz8; gz8[0] = 0; gz8[1] = 0; gz8[2] = 0; gz8[3] = 0;
        gz8[4] = 0; gz8[5] = 0; gz8[6] = 0; gz8[7] = 0;
        __builtin_amdgcn_tensor_load_to_lds(g0, g1, gz4, gz4, gz8, 0);
        __builtin_amdgcn_s_wait_tensorcnt(0);
    }
#else
    for (int i = threadIdx.x; i < TBL_FLOATS; i += BDIM) tbl[i] = ws[i];
#endif
    __syncthreads();

    // Order-preserving prune of the 68 layers against this tile's bbox.
    if (threadIdx.x == 0) {
        int n = 0;
        const float fx0 = (float)X0, fx1 = (float)(X0 + TILE - 1);
        const float fy0 = (float)Y0, fy1 = (float)(Y0 + TILE - 1);
#pragma unroll 1
        for (int s = 0; s < SNUM; ++s) {
            const int po = (b * SNUM + s) * 4;
            const float a = tbl[po], bx = tbl[po + 1], by = tbl[po + 2], mea = tbl[po + 3];
            const bool hit = (mea > 0.0f)
                && (fmaf(a, fx1, bx) > -1.0f) && (fmaf(a, fx0, bx) < 30.0f)
                && (fmaf(a, fy1, by) > -1.0f) && (fmaf(a, fy0, by) < 30.0f);
            if (hit) {
                const int k = s - (s / KNUM) * KNUM;
                plist[n++] = po | ((k * 9) << 16);   // low16: param off, high: sig off
            }
        }
        pcount = n;
    }
    __syncthreads();

    const int tx = threadIdx.x & (TILE - 1);
    const int ty = threadIdx.x >> 5;               // 0..7 (wave32-friendly)
    const float xf = (float)(X0 + tx);
    const int n = pcount;

#pragma unroll
    for (int r = 0; r < 4; ++r) {
        const int Y = Y0 + ty + r * 8;
        const float yf = (float)Y;
        const size_t pix = (size_t)b * 3 * IMG_PLANE + (size_t)Y * HW + (X0 + tx);
        float c0 = img[pix];
        float c1 = img[pix + IMG_PLANE];
        float c2 = img[pix + 2 * IMG_PLANE];
        for (int i = 0; i < n; ++i) {
            const int e = plist[i];
            const float* Pp = &tbl[e & 0xFFFF];
            const float a = Pp[0], bx = Pp[1], by = Pp[2], mea = Pp[3];
            const float ix = fmaf(a, xf, bx);
            const float iy = fmaf(a, yf, by);
            if (!(ix > -1.0f && ix < 30.0f && iy > -1.0f && iy < 30.0f)) continue;
            const float ix0f = floorf(ix), iy0f = floorf(iy);
            const float wx = ix - ix0f, wy = iy - iy0f;
            const int x0 = (int)ix0f, y0 = (int)iy0f;
            const float* sg = &tbl[PARAM_FLOATS + (e >> 16)];
            const float fvx0 = ((unsigned)x0        < 30u) ? 1.0f : 0.0f;
            const float fvx1 = ((unsigned)(x0 + 1)  < 30u) ? 1.0f : 0.0f;
            const float fvy0 = ((unsigned)y0        < 30u) ? 1.0f : 0.0f;
            const float fvy1 = ((unsigned)(y0 + 1)  < 30u) ? 1.0f : 0.0f;
            const int cx0 = (x0 < 0) ? 0 : x0 / 10;       // nearest-upsample fold
            const int cx1 = (x0 + 1) / 10;
            const int ry0 = ((y0 < 0) ? 0 : y0 / 10) * 3;
            const int ry1 = ((y0 + 1) / 10) * 3;
            const float t00 = sg[ry0 + cx0] * (fvx0 * fvy0);
            const float t01 = sg[ry0 + cx1] * (fvx1 * fvy0);
            const float t10 = sg[ry1 + cx0] * (fvx0 * fvy1);
            const float t11 = sg[ry1 + cx1] * (fvx1 * fvy1);
            const float wx0 = 1.0f - wx, wy0 = 1.0f - wy;
            const float p = (t00 * wx0 + t01 * wx) * wy0 + (t10 * wx0 + t11 * wx) * wy;
            const float m = (fvx0 * wx0 + fvx1 * wx) * (fvy0 * wy0 + fvy1 * wy);
            const float me = m * mea;
            const float om = 1.0f - me;
            const float pme = p * me;
            c0 = fmaf(c0, om, pme);
            c1 = fmaf(c1, om, pme);
            c2 = fmaf(c2, om, pme);
        }
        out[pix]                 = c0;
        out[pix + IMG_PLANE]     = c1;
        out[pix + 2 * IMG_PLANE] = c2;
    }
}

// ---------------------------------------------------------------------------
extern "C" void kernel_launch(void* const* d_in, const int* in_sizes, int n_in,
                              void* d_out, int out_size, void* d_ws, size_t ws_size,
                              hipStream_t stream) {
    (void)in_sizes; (void)n_in; (void)out_size; (void)ws_size;
    const float* images = (const float*)d_in[0];
    const float* tkp    = (const float*)d_in[1];
    const float* tbb    = (const float*)d_in[2];
    const float* tex    = (const float*)d_in[3];
    const float* slog   = (const float*)d_in[4];
    const float* alog   = (const float*)d_in[5];
    float* out = (float*)d_out;
    float* ws  = (float*)d_ws;

    pap_prep<<<1, BDIM, 0, stream>>>(tkp, tbb, tex, slog, alog, ws, out + IMG_ELEMS);

    dim3 grid(HW / TILE, HW / TILE, BNUM);
    pap_blend<<<grid, BDIM, 0, stream>>>(images, ws, out);
}